// changedGATConv_1700807049271
// MI455X (gfx1250) — compile-verified
//
#include <hip/hip_runtime.h>
#include <math.h>

#define N_NODES   50000
#define N_EDGES   800000
#define IN_FEATS  256
#define OUT_FEATS 64
#define NUM_HEADS 4
#define EDGE_FEATS 64
#define NUM_ETYPES 8
#define NEG_SLOPE 0.2f
#define HD (NUM_HEADS * OUT_FEATS)   /* 256 */

typedef __attribute__((ext_vector_type(2))) float v2f;
typedef __attribute__((ext_vector_type(8))) float v8f;

// ---------------------------------------------------------------- init
__global__ void k_init(float* rst, unsigned* mbits, float* sbuf) {
    int i = blockIdx.x * 256 + threadIdx.x;
    if (i < N_NODES * HD) rst[i] = 0.0f;
    if (i < N_NODES * NUM_HEADS) {
        mbits[i] = 0xff800000u;   // -inf
        sbuf[i]  = 0.0f;
    }
}

// ------------------------------------------------- ee table: [8][4]
// proj[t][j] = sum_k edge_emb[t][k] * W_e[k][j]   (j in [0,256))
// ee[t][h]   = sum_f proj[t][h*64+f] * attn_e[h][f]   (attn_e flat index == j)
__global__ void k_ee(const float* __restrict__ edge_emb,
                     const float* __restrict__ W_e,
                     const float* __restrict__ attn_e,
                     float* __restrict__ ee) {
    __shared__ float lds[256];
    int t = blockIdx.x;
    int j = threadIdx.x;
    float acc = 0.0f;
    for (int k = 0; k < EDGE_FEATS; ++k)
        acc += edge_emb[t * EDGE_FEATS + k] * W_e[k * HD + j];
    lds[j] = acc * attn_e[j];
    __syncthreads();
    if (j < NUM_HEADS) {
        float s = 0.0f;
        for (int f = 0; f < OUT_FEATS; ++f) s += lds[j * OUT_FEATS + f];
        ee[t * NUM_HEADS + j] = s;
    }
}

// ------------------------------------------------- ft = feat @ W_fc (WMMA f32)
// Block: 256 threads = 8 waves. Block owns 16 rows; wave w owns cols [32w, 32w+32).
// A (16x4 f32): lanes 0-15 hold M, VGPR0 = K{0|2}, VGPR1 = K{1|3} (ISA 7.12.2).
// B (4x16 f32): VGPR0 lanes0-15 K=0 / lanes16-31 K=2, VGPR1 K=1/K=3, N = lane%16.
// C/D (16x16 f32): VGPR j: lanes0-15 M=j, lanes16-31 M=j+8, N = lane%16.
__global__ void __launch_bounds__(256) k_gemm(const float* __restrict__ feat,
                                              const float* __restrict__ W,
                                              float* __restrict__ ft) {
    const int tid  = threadIdx.x;
    const int wave = tid >> 5;
    const int lane = tid & 31;
    const int half = lane >> 4;
    const int l16  = lane & 15;
    const int m0   = blockIdx.x * 16;
    const int n0   = wave * 32;

    const float* Arow = feat + (m0 + l16) * IN_FEATS;
    v8f c0 = {};
    v8f c1 = {};
    for (int k = 0; k < IN_FEATS; k += 4) {
        const int ka = k + 2 * half;
        v2f a, b0, b1;
        a[0]  = Arow[ka];
        a[1]  = Arow[ka + 1];
        b0[0] = W[ka * HD + n0 + l16];
        b0[1] = W[(ka + 1) * HD + n0 + l16];
        b1[0] = W[ka * HD + n0 + 16 + l16];
        b1[1] = W[(ka + 1) * HD + n0 + 16 + l16];
        c0 = __builtin_amdgcn_wmma_f32_16x16x4_f32(false, a, false, b0, (short)0, c0, false, false);
        c1 = __builtin_amdgcn_wmma_f32_16x16x4_f32(false, a, false, b1, (short)0, c1, false, false);
    }
#pragma unroll
    for (int j = 0; j < 8; ++j) {
        const int row = m0 + j + 8 * half;
        ft[row * HD + n0 + l16]      = c0[j];
        ft[row * HD + n0 + 16 + l16] = c1[j];
    }
}

// ------------------------------------------------- el/er: [N,H] attention dots
__global__ void k_elr(const float* __restrict__ ft,
                      const float* __restrict__ attn_l,
                      const float* __restrict__ attn_r,
                      float* __restrict__ el, float* __restrict__ er) {
    int i = blockIdx.x * 256 + threadIdx.x;
    if (i >= N_NODES * NUM_HEADS) return;
    const int n = i >> 2, h = i & 3;
    const float* f  = ft + n * HD + h * OUT_FEATS;
    const float* al = attn_l + h * OUT_FEATS;
    const float* ar = attn_r + h * OUT_FEATS;
    float sl = 0.0f, sr = 0.0f;
    for (int d = 0; d < OUT_FEATS; ++d) {
        const float v = f[d];
        sl += v * al[d];
        sr += v * ar[d];
    }
    el[i] = sl;
    er[i] = sr;
}

__device__ __forceinline__ float leaky(float e) { return e > 0.0f ? e : NEG_SLOPE * e; }

__device__ __forceinline__ void atomicMaxF(float* addr, float v) {
    if (v >= 0.0f) atomicMax((int*)addr, __float_as_int(v));
    else           atomicMin((unsigned int*)addr, __float_as_uint(v));
}

// ------------------------------------------------- edge pass 1: segment max
__global__ void k_edge_max(const int* __restrict__ efeat,
                           const int* __restrict__ src,
                           const int* __restrict__ dst,
                           const float* __restrict__ el,
                           const float* __restrict__ er,
                           const float* __restrict__ ee,
                           float* __restrict__ mbuf) {
    int e = blockIdx.x * 256 + threadIdx.x;
    if (e >= N_EDGES) return;
    const int s = src[e], d = dst[e], t = efeat[e];
#pragma unroll
    for (int h = 0; h < NUM_HEADS; ++h) {
        const float x = leaky(el[s * 4 + h] + er[d * 4 + h] + ee[t * 4 + h]);
        atomicMaxF(&mbuf[d * 4 + h], x);
    }
}

// ------------------------------------------------- edge pass 2: exp + segment sum
__global__ void k_edge_exp(const int* __restrict__ efeat,
                           const int* __restrict__ src,
                           const int* __restrict__ dst,
                           const float* __restrict__ el,
                           const float* __restrict__ er,
                           const float* __restrict__ ee,
                           const float* __restrict__ mbuf,
                           float* __restrict__ sbuf,
                           float* __restrict__ a_out) {
    int e = blockIdx.x * 256 + threadIdx.x;
    if (e >= N_EDGES) return;
    const int s = src[e], d = dst[e], t = efeat[e];
#pragma unroll
    for (int h = 0; h < NUM_HEADS; ++h) {
        const float x  = leaky(el[s * 4 + h] + er[d * 4 + h] + ee[t * 4 + h]);
        const float ex = expf(x - mbuf[d * 4 + h]);
        a_out[e * 4 + h] = ex;
        atomicAdd(&sbuf[d * 4 + h], ex);
    }
}

// ------------------------------------------------- edge pass 3: normalize + scatter
// One wave per edge. Lane l covers ft elements j = l + 32*i (i<8); head = i/2.
__global__ void __launch_bounds__(256) k_aggregate(const int* __restrict__ src,
                                                   const int* __restrict__ dst,
                                                   const float* __restrict__ ft,
                                                   const float* __restrict__ sbuf,
                                                   float* __restrict__ a_out,
                                                   float* __restrict__ rst) {
    const int wave = threadIdx.x >> 5;
    const int lane = threadIdx.x & 31;
    const int e = blockIdx.x * 8 + wave;
    if (e >= N_EDGES) return;
    const int s = src[e], d = dst[e];

    float a_local = 0.0f;
    if (lane < NUM_HEADS) {
        a_local = a_out[e * 4 + lane] / sbuf[d * 4 + lane];
        a_out[e * 4 + lane] = a_local;   // final normalized attention output
    }

    const float* f = ft + s * HD;
    float* r = rst + d * HD;
#pragma unroll
    for (int i = 0; i < 8; ++i) {
        const int j = lane + 32 * i;
        const float av = __shfl(a_local, i >> 1, 32);
        atomicAdd(&r[j], f[j] * av);
    }
}

// ================================================================ launch
extern "C" void kernel_launch(void* const* d_in, const int* in_sizes, int n_in,
                              void* d_out, int out_size, void* d_ws, size_t ws_size,
                              hipStream_t stream) {
    const float* feat     = (const float*)d_in[0];
    const int*   e_feat   = (const int*)  d_in[1];
    const int*   src      = (const int*)  d_in[2];
    const int*   dst      = (const int*)  d_in[3];
    const float* W_fc     = (const float*)d_in[4];
    const float* W_e      = (const float*)d_in[5];
    const float* edge_emb = (const float*)d_in[6];
    const float* attn_l   = (const float*)d_in[7];
    const float* attn_r   = (const float*)d_in[8];
    const float* attn_e   = (const float*)d_in[9];

    float* rst   = (float*)d_out;                       // [N, H, D]
    float* a_out = rst + (size_t)N_NODES * HD;          // [E, H]

    float* ws   = (float*)d_ws;
    float* ft   = ws;                                   // N*HD  = 12.8M floats
    float* el   = ft + (size_t)N_NODES * HD;            // N*H
    float* er   = el + N_NODES * NUM_HEADS;
    float* mbuf = er + N_NODES * NUM_HEADS;
    float* sbuf = mbuf + N_NODES * NUM_HEADS;
    float* ee   = sbuf + N_NODES * NUM_HEADS;           // 32 floats

    k_init<<<(N_NODES * HD + 255) / 256, 256, 0, stream>>>(rst, (unsigned*)mbuf, sbuf);
    k_ee<<<NUM_ETYPES, 256, 0, stream>>>(edge_emb, W_e, attn_e, ee);
    k_gemm<<<N_NODES / 16, 256, 0, stream>>>(feat, W_fc, ft);
    k_elr<<<(N_NODES * NUM_HEADS + 255) / 256, 256, 0, stream>>>(ft, attn_l, attn_r, el, er);
    k_edge_max<<<(N_EDGES + 255) / 256, 256, 0, stream>>>(e_feat, src, dst, el, er, ee, mbuf);
    k_edge_exp<<<(N_EDGES + 255) / 256, 256, 0, stream>>>(e_feat, src, dst, el, er, ee, mbuf, sbuf, a_out);
    k_aggregate<<<N_EDGES / 8, 256, 0, stream>>>(src, dst, ft, sbuf, a_out, rst);
}